// GraphTransformerEncoderLayer_46480136077660
// MI455X (gfx1250) — compile-verified
//
#include <hip/hip_runtime.h>
#include <hip/hip_bf16.h>

#define NN 100000
#define FD 128
#define NH 8
#define NC 16
#define NE 1600000
#define FFD 512
#define NEG_SLOPE 0.2f
#define LN_EPS 1e-5f

typedef __attribute__((ext_vector_type(2))) float v2f;
typedef __attribute__((ext_vector_type(8))) float v8f;

__device__ __forceinline__ v8f wmma_f32(v2f a, v2f b, v8f c) {
    // D(16x16,f32) = A(16x4,f32) * B(4x16,f32) + C
    return __builtin_amdgcn_wmma_f32_16x16x4_f32(
        false, a, false, b, (short)0, c, false, false);
}

// order-preserving float<->uint maps for atomicMax-based segment max
__device__ __forceinline__ unsigned f2o(float f) {
    unsigned u = __float_as_uint(f);
    return (u & 0x80000000u) ? ~u : (u | 0x80000000u);
}
__device__ __forceinline__ float o2f(unsigned o) {
    unsigned u = (o & 0x80000000u) ? (o & 0x7FFFFFFFu) : ~o;
    return __uint_as_float(u);
}

// ---------------- init: zero m / denom / h accumulator ----------------
__global__ void k_init(unsigned* __restrict__ p, long long n) {
    long long i = (long long)blockIdx.x * blockDim.x + threadIdx.x;
    long long stride = (long long)gridDim.x * blockDim.x;
    for (; i < n; i += stride) p[i] = 0u;
}

// ---------------- fused xl = x@Wl+bl, xr = x@Wr+br --------------------
// grid 6250, block 128 (4 waves). Waves 0-1 -> Wl cols, waves 2-3 -> Wr cols.
// Each wave: 4 col-tiles interleaved in one fully-unrolled K loop.
__global__ void k_lin2(const float* __restrict__ x,
                       const float* __restrict__ Wl, const float* __restrict__ bl,
                       const float* __restrict__ Wr, const float* __restrict__ br,
                       float* __restrict__ xl, float* __restrict__ xr) {
    __shared__ float xa[16 * 132];
    const int m0 = blockIdx.x * 16;
    const int tid = threadIdx.x;
    const int wave = tid >> 5;
    const int lane = tid & 31;
    const int half = lane >> 4;
    const int l15 = lane & 15;

    for (int i = tid; i < 16 * 32; i += 128) {
        int r = i >> 5, c = (i & 31) * 4;
        const float4 v = *(const float4*)(x + (size_t)(m0 + r) * FD + c);
        float* p = &xa[r * 132 + c];
        p[0] = v.x; p[1] = v.y; p[2] = v.z; p[3] = v.w;
    }
    __syncthreads();

    const float* W   = (wave < 2) ? Wl : Wr;
    const float* bia = (wave < 2) ? bl : br;
    float*       op  = (wave < 2) ? xl : xr;
    const int c0 = (wave & 1) * 64 + l15;       // 4 tiles: c0, c0+16, c0+32, c0+48
    const float* Wp = W + c0;                    // single base, immediate offsets
    const float* As = &xa[l15 * 132];

    v8f a0 = {}, a1 = {}, a2 = {}, a3 = {};
#pragma unroll
    for (int k = 0; k < FD; k += 4) {
        const int ka = k + half * 2;
        const v2f a = *(const v2f*)(As + ka);
        v2f b0, b1, b2, b3;
        b0.x = Wp[ka * FD];       b0.y = Wp[(ka + 1) * FD];
        b1.x = Wp[ka * FD + 16];  b1.y = Wp[(ka + 1) * FD + 16];
        b2.x = Wp[ka * FD + 32];  b2.y = Wp[(ka + 1) * FD + 32];
        b3.x = Wp[ka * FD + 48];  b3.y = Wp[(ka + 1) * FD + 48];
        a0 = wmma_f32(a, b0, a0);
        a1 = wmma_f32(a, b1, a1);
        a2 = wmma_f32(a, b2, a2);
        a3 = wmma_f32(a, b3, a3);
    }

    const float bv0 = bia[c0], bv1 = bia[c0 + 16], bv2 = bia[c0 + 32], bv3 = bia[c0 + 48];
    float* po = op + (size_t)(m0 + half * 8) * FD + c0;
#pragma unroll
    for (int i = 0; i < 8; ++i) {
        po[i * FD]      = a0[i] + bv0;
        po[i * FD + 16] = a1[i] + bv1;
        po[i * FD + 32] = a2[i] + bv2;
        po[i * FD + 48] = a3[i] + bv3;
    }
}

// ---------------- edge scores + segment max ---------------------------
__global__ void k_score(const int* __restrict__ ei,
                        const float* __restrict__ xl, const float* __restrict__ xr,
                        const float* __restrict__ att,
                        float* __restrict__ ebuf, unsigned* __restrict__ m_u) {
    long long idx = (long long)blockIdx.x * blockDim.x + threadIdx.x;
    if (idx >= (long long)NE * NH) return;
    const int e = (int)(idx >> 3);
    const int h = (int)(idx & 7);
    const int src = ei[e];
    const int dst = ei[NE + e];
    const float4* pj = (const float4*)(xl + (size_t)src * FD + h * NC);
    const float4* pi = (const float4*)(xr + (size_t)dst * FD + h * NC);
    const float4* pa = (const float4*)(att + h * NC);
    float s = 0.f;
#pragma unroll
    for (int q = 0; q < 4; ++q) {
        const float4 vj = pj[q];
        const float4 vi = pi[q];
        const float4 va = pa[q];
        float t;
        t = vi.x + vj.x; s += va.x * ((t > 0.f) ? t : t * NEG_SLOPE);
        t = vi.y + vj.y; s += va.y * ((t > 0.f) ? t : t * NEG_SLOPE);
        t = vi.z + vj.z; s += va.z * ((t > 0.f) ? t : t * NEG_SLOPE);
        t = vi.w + vj.w; s += va.w * ((t > 0.f) ? t : t * NEG_SLOPE);
    }
    ebuf[idx] = s;
    atomicMax(m_u + (size_t)dst * NH + h, f2o(s));
}

// ---------------- exp(e - m) + segment sum ----------------------------
__global__ void k_exp(const int* __restrict__ ei,
                      float* __restrict__ ebuf,
                      const unsigned* __restrict__ m_u,
                      float* __restrict__ denom) {
    long long idx = (long long)blockIdx.x * blockDim.x + threadIdx.x;
    if (idx >= (long long)NE * NH) return;
    const int e = (int)(idx >> 3);
    const int h = (int)(idx & 7);
    const int dst = ei[NE + e];
    const float m = o2f(m_u[(size_t)dst * NH + h]);
    const float ex = expf(ebuf[idx] - m);
    ebuf[idx] = ex;
    atomicAdd(denom + (size_t)dst * NH + h, ex);
}

// ---------------- aggregation: h[dst] += xl[src] * alpha --------------
// 1 wave per edge, 4 channels per lane.
__global__ void k_agg(const int* __restrict__ ei,
                      const float* __restrict__ xl,
                      const float* __restrict__ ebuf,
                      const float* __restrict__ denom,
                      float* __restrict__ hbuf) {
    long long tid = (long long)blockIdx.x * blockDim.x + threadIdx.x;
    const int e = (int)(tid >> 5);
    if (e >= NE) return;
    const int sub = (int)(tid & 31);
    const int f0 = sub * 4;
    const int h = sub >> 2;
    const int src = ei[e];
    const int dst = ei[NE + e];
    const float alpha = ebuf[(size_t)e * NH + h] / denom[(size_t)dst * NH + h];
    const float4 xj = *(const float4*)(xl + (size_t)src * FD + f0);
    float* po = hbuf + (size_t)dst * FD + f0;
    atomicAdd(po + 0, xj.x * alpha);
    atomicAdd(po + 1, xj.y * alpha);
    atomicAdd(po + 2, xj.z * alpha);
    atomicAdd(po + 3, xj.w * alpha);
}

// ---------------- + bias_gat, LayerNorm1 ------------------------------
__global__ void k_bias_ln1(float* __restrict__ hbuf,
                           const float* __restrict__ bias_gat,
                           const float* __restrict__ g1, const float* __restrict__ bn1,
                           float* __restrict__ ln1) {
    __shared__ float2 red[128];
    const int row = blockIdx.x;
    const int t = threadIdx.x;
    float v = hbuf[(size_t)row * FD + t] + bias_gat[t];
    hbuf[(size_t)row * FD + t] = v;
    red[t] = make_float2(v, v * v);
    __syncthreads();
    for (int s = 64; s > 0; s >>= 1) {
        if (t < s) {
            red[t].x += red[t + s].x;
            red[t].y += red[t + s].y;
        }
        __syncthreads();
    }
    const float mu = red[0].x * (1.0f / FD);
    const float var = red[0].y * (1.0f / FD) - mu * mu;
    const float rs = rsqrtf(var + LN_EPS);
    ln1[(size_t)row * FD + t] = (v - mu) * rs * g1[t] + bn1[t];
}

// ---------------- FFN1: a1 = silu(ln1 @ W1 + bW1) ---------------------
// grid 6250, block 256 (8 waves), each wave 4 col-tiles interleaved.
__global__ void k_ffn1(const float* __restrict__ ln1,
                       const float* __restrict__ W1, const float* __restrict__ bW1,
                       float* __restrict__ a1) {
    __shared__ float xa[16 * 132];
    const int m0 = blockIdx.x * 16;
    const int tid = threadIdx.x;
    const int wave = tid >> 5;
    const int lane = tid & 31;
    const int half = lane >> 4;
    const int l15 = lane & 15;

    for (int i = tid; i < 16 * 32; i += 256) {
        int r = i >> 5, c = (i & 31) * 4;
        const float4 v = *(const float4*)(ln1 + (size_t)(m0 + r) * FD + c);
        float* p = &xa[r * 132 + c];
        p[0] = v.x; p[1] = v.y; p[2] = v.z; p[3] = v.w;
    }
    __syncthreads();

    const int c0 = wave * 64 + l15;              // 4 tiles: +0,+16,+32,+48
    const float* Wp = W1 + c0;
    const float* As = &xa[l15 * 132];

    v8f a0 = {}, a1v = {}, a2 = {}, a3 = {};
#pragma unroll
    for (int k = 0; k < FD; k += 4) {
        const int ka = k + half * 2;
        const v2f a = *(const v2f*)(As + ka);
        v2f b0, b1, b2, b3;
        b0.x = Wp[ka * FFD];       b0.y = Wp[(ka + 1) * FFD];
        b1.x = Wp[ka * FFD + 16];  b1.y = Wp[(ka + 1) * FFD + 16];
        b2.x = Wp[ka * FFD + 32];  b2.y = Wp[(ka + 1) * FFD + 32];
        b3.x = Wp[ka * FFD + 48];  b3.y = Wp[(ka + 1) * FFD + 48];
        a0  = wmma_f32(a, b0, a0);
        a1v = wmma_f32(a, b1, a1v);
        a2  = wmma_f32(a, b2, a2);
        a3  = wmma_f32(a, b3, a3);
    }

    const float bv0 = bW1[c0], bv1 = bW1[c0 + 16], bv2 = bW1[c0 + 32], bv3 = bW1[c0 + 48];
    float* po = a1 + (size_t)(m0 + half * 8) * FFD + c0;
#pragma unroll
    for (int i = 0; i < 8; ++i) {
        float v;
        v = a0[i] + bv0;  po[i * FFD]      = v / (1.f + expf(-v));
        v = a1v[i] + bv1; po[i * FFD + 16] = v / (1.f + expf(-v));
        v = a2[i] + bv2;  po[i * FFD + 32] = v / (1.f + expf(-v));
        v = a3[i] + bv3;  po[i * FFD + 48] = v / (1.f + expf(-v));
    }
}

// ---------------- FFN2 + residual + LayerNorm2 -> out -----------------
// grid 6250, block 128 (4 waves), each wave 2 col-tiles; K = 512 staged in LDS.
__global__ void k_ffn2(const float* __restrict__ a1,
                       const float* __restrict__ W2, const float* __restrict__ bW2,
                       const float* __restrict__ hbuf,
                       const float* __restrict__ g2, const float* __restrict__ bn2,
                       float* __restrict__ out) {
    __shared__ float sm[16 * 520];   // K tile 16x512 (stride 520); reused as out tile 16x132
    __shared__ float ps[16][8][2];
    __shared__ float stat[16][2];
    const int m0 = blockIdx.x * 16;
    const int tid = threadIdx.x;
    const int wave = tid >> 5;
    const int lane = tid & 31;
    const int half = lane >> 4;
    const int l15 = lane & 15;

    for (int i = tid; i < 2048; i += 128) {          // 16 rows * 128 float4
        int r = i >> 7, c = (i & 127) * 4;
        const float4 v = *(const float4*)(a1 + (size_t)(m0 + r) * FFD + c);
        float* p = &sm[r * 520 + c];
        p[0] = v.x; p[1] = v.y; p[2] = v.z; p[3] = v.w;
    }
    __syncthreads();

    const int c0 = wave * 32 + l15;              // 2 tiles: +0, +16
    const float* Wp = W2 + c0;
    const float* As = &sm[l15 * 520];

    v8f acc0 = {}, acc1 = {};
#pragma unroll
    for (int k = 0; k < FFD; k += 4) {
        const int ka = k + half * 2;
        const v2f a = *(const v2f*)(As + ka);
        v2f b0, b1;
        b0.x = Wp[ka * FD];       b0.y = Wp[(ka + 1) * FD];
        b1.x = Wp[ka * FD + 16];  b1.y = Wp[(ka + 1) * FD + 16];
        acc0 = wmma_f32(a, b0, acc0);
        acc1 = wmma_f32(a, b1, acc1);
    }
    __syncthreads();   // K tile fully consumed; reuse sm as 16x132 output tile

    const float bv0 = bW2[c0], bv1 = bW2[c0 + 16];
#pragma unroll
    for (int i = 0; i < 8; ++i) {
        const int r = i + half * 8;
        sm[r * 132 + c0]      = acc0[i] + bv0 + hbuf[(size_t)(m0 + r) * FD + c0];
        sm[r * 132 + c0 + 16] = acc1[i] + bv1 + hbuf[(size_t)(m0 + r) * FD + c0 + 16];
    }
    __syncthreads();

    {   // partial LN stats: 8 threads per row, 16 cols each
        const int r = tid >> 3, part = tid & 7;
        float s = 0.f, s2 = 0.f;
        for (int c = part * 16; c < part * 16 + 16; ++c) {
            const float v = sm[r * 132 + c];
            s += v; s2 += v * v;
        }
        ps[r][part][0] = s; ps[r][part][1] = s2;
    }
    __syncthreads();
    if (tid < 16) {
        float s = 0.f, s2 = 0.f;
        for (int p = 0; p < 8; ++p) { s += ps[tid][p][0]; s2 += ps[tid][p][1]; }
        const float mu = s * (1.0f / FD);
        const float var = s2 * (1.0f / FD) - mu * mu;
        stat[tid][0] = mu;
        stat[tid][1] = rsqrtf(var + LN_EPS);
    }
    __syncthreads();
    for (int i = tid; i < 2048; i += 128) {
        const int r = i >> 7, c = i & 127;
        const float v = (sm[r * 132 + c] - stat[r][0]) * stat[r][1] * g2[c] + bn2[c];
        out[(size_t)(m0 + r) * FD + c] = v;
    }
}

extern "C" void kernel_launch(void* const* d_in, const int* in_sizes, int n_in,
                              void* d_out, int out_size, void* d_ws, size_t ws_size,
                              hipStream_t stream) {
    const float* x        = (const float*)d_in[0];
    const int*   ei       = (const int*)  d_in[1];
    const float* Wl       = (const float*)d_in[2];
    const float* bl       = (const float*)d_in[3];
    const float* Wr       = (const float*)d_in[4];
    const float* br       = (const float*)d_in[5];
    const float* att      = (const float*)d_in[6];
    const float* bias_gat = (const float*)d_in[7];
    const float* g1       = (const float*)d_in[8];
    const float* bn1      = (const float*)d_in[9];
    const float* W1       = (const float*)d_in[10];
    const float* bW1      = (const float*)d_in[11];
    const float* W2       = (const float*)d_in[12];
    const float* bW2      = (const float*)d_in[13];
    const float* g2       = (const float*)d_in[14];
    const float* bn2      = (const float*)d_in[15];
    float* out = (float*)d_out;

    // workspace carve-up (32-bit words)
    float* ws = (float*)d_ws;
    const size_t NF = (size_t)NN * FD;          // 12.8M
    float*    xl    = ws;
    float*    xr    = xl + NF;
    float*    ebuf  = xr + NF;                  // NE*NH = 12.8M
    unsigned* m_u   = (unsigned*)(ebuf + (size_t)NE * NH);
    float*    denom = (float*)(m_u + (size_t)NN * NH);
    float*    hbuf  = denom + (size_t)NN * NH;
    float*    ln1   = hbuf + NF;
    float*    a1    = ln1 + NF;                 // NN*FFD = 51.2M

    // 1) zero m_u + denom + hbuf (contiguous)
    const long long zn = (long long)NN * NH * 2 + (long long)NF;
    k_init<<<4096, 256, 0, stream>>>(m_u, zn);

    // 2) node linear transforms (WMMA)
    k_lin2<<<NN / 16, 128, 0, stream>>>(x, Wl, bl, Wr, br, xl, xr);

    // 3) edge scores + segment max
    const int sthreads = 256;
    const long long eh = (long long)NE * NH;
    k_score<<<(unsigned)((eh + sthreads - 1) / sthreads), sthreads, 0, stream>>>(
        ei, xl, xr, att, ebuf, m_u);

    // 4) exp + segment sum
    k_exp<<<(unsigned)((eh + sthreads - 1) / sthreads), sthreads, 0, stream>>>(
        ei, ebuf, m_u, denom);

    // 5) scatter-add aggregation (wave per edge)
    const long long at = (long long)NE * 32;
    k_agg<<<(unsigned)((at + 255) / 256), 256, 0, stream>>>(ei, xl, ebuf, denom, hbuf);

    // 6) + bias_gat, LayerNorm1
    k_bias_ln1<<<NN, 128, 0, stream>>>(hbuf, bias_gat, g1, bn1, ln1);

    // 7) FFN up-projection + silu (WMMA)
    k_ffn1<<<NN / 16, 256, 0, stream>>>(ln1, W1, bW1, a1);

    // 8) FFN down-projection + residual + LayerNorm2 (WMMA)
    k_ffn2<<<NN / 16, 128, 0, stream>>>(a1, W2, bW2, hbuf, g2, bn2, out);
}